// POI2Region_28965259444228
// MI455X (gfx1250) — compile-verified
//
#include <hip/hip_runtime.h>
#include <hip/hip_bf16.h>
#include <math.h>

typedef float v2f __attribute__((ext_vector_type(2)));
typedef float v8f __attribute__((ext_vector_type(8)));

#define D128 128
#define NH 4

// ---------------------------------------------------------------- init
__global__ void k_init(int* counts, int* cursor, float* deg, float* acc, int R) {
  int i = blockIdx.x * blockDim.x + threadIdx.x;
  if (i < R) { counts[i] = 0; cursor[i] = 0; deg[i] = 1.0f; /* self loop */ }
  if (i < R * D128) acc[i] = 0.0f;
}

// ---------------------------------------------------------------- prep: q, A, c
__global__ void k_prep(const float* __restrict__ S, const float* __restrict__ Wq,
                       const float* __restrict__ bq, const float* __restrict__ Wk,
                       const float* __restrict__ bk,
                       float* __restrict__ qv, float* __restrict__ Av, float* __restrict__ cv) {
  __shared__ float qs[D128];
  int t = threadIdx.x;                       // 128 threads
  float acc = bq[t];
  for (int d = 0; d < D128; ++d) acc = fmaf(S[d], Wq[t * D128 + d], acc);
  qs[t] = acc;
  qv[t] = acc;
  __syncthreads();
  const float scale = 0.08838834764831845f;  // 1/sqrt(128)
  for (int h = 0; h < NH; ++h) {
    float a = 0.0f;
    for (int j = 0; j < 32; ++j) a = fmaf(qs[h * 32 + j], Wk[(h * 32 + j) * D128 + t], a);
    Av[h * D128 + t] = a * scale;
  }
  if (t < NH) {
    float cc = 0.0f;
    for (int j = 0; j < 32; ++j) cc = fmaf(qs[t * 32 + j], bk[t * 32 + j], cc);
    cv[t] = cc * scale;
  }
}

// ---------------------------------------------------------------- pass 1: scores + histogram
__global__ void k_scores(const float* __restrict__ x, const int* __restrict__ zone,
                         const float* __restrict__ Av, const float* __restrict__ cv,
                         float* __restrict__ scores, int* __restrict__ counts, int N) {
  __shared__ float As[NH * D128];
  __shared__ float cs[NH];
  for (int i = threadIdx.x; i < NH * D128; i += blockDim.x) As[i] = Av[i];
  if (threadIdx.x < NH) cs[threadIdx.x] = cv[threadIdx.x];
  __syncthreads();
  int n = blockIdx.x * blockDim.x + threadIdx.x;
  if (n >= N) return;
  const float4* xr = (const float4*)(x + (size_t)n * D128);
  float s[NH] = {cs[0], cs[1], cs[2], cs[3]};
  for (int i = 0; i < 32; ++i) {
    float4 xv = xr[i];
#pragma unroll
    for (int h = 0; h < NH; ++h) {
      const float* a = As + h * D128 + 4 * i;
      s[h] = fmaf(xv.x, a[0], fmaf(xv.y, a[1], fmaf(xv.z, a[2], fmaf(xv.w, a[3], s[h]))));
    }
  }
#pragma unroll
  for (int h = 0; h < NH; ++h) scores[(size_t)n * NH + h] = s[h];
  atomicAdd(&counts[zone[n]], 1);
}

// ---------------------------------------------------------------- exclusive scan (R <= 2048 fast path)
__global__ void k_scan(const int* __restrict__ counts, int* __restrict__ offsets, int R) {
  __shared__ int bufA[1024];
  __shared__ int bufB[1024];
  int t = threadIdx.x;
  if (R > 2048) {
    if (t == 0) { int a = 0; for (int r = 0; r < R; ++r) { offsets[r] = a; a += counts[r]; } offsets[R] = a; }
    return;
  }
  int c0 = (2 * t     < R) ? counts[2 * t]     : 0;
  int c1 = (2 * t + 1 < R) ? counts[2 * t + 1] : 0;
  bufA[t] = c0 + c1;
  __syncthreads();
  int* src = bufA; int* dst = bufB;
  for (int d = 1; d < 1024; d <<= 1) {
    int v = src[t];
    if (t >= d) v += src[t - d];
    __syncthreads();
    dst[t] = v;
    __syncthreads();
    int* tmp = src; src = dst; dst = tmp;
  }
  int excl = (t == 0) ? 0 : src[t - 1];
  if (2 * t     < R) offsets[2 * t]     = excl;
  if (2 * t + 1 < R) offsets[2 * t + 1] = excl + c0;
  if (t == 1023) offsets[R] = src[1023];
}

// ---------------------------------------------------------------- scatter point ids into CSR
__global__ void k_scatter(const int* __restrict__ zone, const int* __restrict__ offsets,
                          int* __restrict__ cursor, int* __restrict__ idx, int N) {
  int n = blockIdx.x * blockDim.x + threadIdx.x;
  if (n >= N) return;
  int r = zone[n];
  int pos = atomicAdd(&cursor[r], 1);
  idx[offsets[r] + pos] = n;
}

// ---------------------------------------------------------------- per-region: max, exp, denom, y = sum e*x
#define K5T 512
__global__ void k_region(const float* __restrict__ x, const float* __restrict__ scores,
                         const int* __restrict__ idx, const int* __restrict__ offsets,
                         float* __restrict__ y, float* __restrict__ denom) {
  int r = blockIdx.x;
  int tid = threadIdx.x;
  int off = offsets[r];
  int cnt = offsets[r + 1] - off;
  __shared__ float mfin[NH];
  __shared__ float e_s[32 * NH];
  __shared__ int   n_s[32];
  __shared__ float red[16 * NH];

  if (cnt == 0) {                      // never happens statistically; keep buffers sane
    y[(size_t)r * (NH * D128) + tid] = 0.0f;
    if (tid < NH) denom[r * NH + tid] = 1.0f;
    return;
  }
  // phase A: per-head max
  float lmax[NH] = {-3.4e38f, -3.4e38f, -3.4e38f, -3.4e38f};
  for (int p = tid; p < cnt; p += K5T) {
    int n = idx[off + p];
    const float* s = scores + (size_t)n * NH;
#pragma unroll
    for (int h = 0; h < NH; ++h) lmax[h] = fmaxf(lmax[h], s[h]);
  }
#pragma unroll
  for (int h = 0; h < NH; ++h)
    for (int d = 16; d; d >>= 1) lmax[h] = fmaxf(lmax[h], __shfl_xor(lmax[h], d));
  int wid = tid >> 5;
  if ((tid & 31) == 0)
#pragma unroll
    for (int h = 0; h < NH; ++h) red[wid * NH + h] = lmax[h];
  __syncthreads();
  if (tid < NH) {
    float m = -3.4e38f;
    for (int w = 0; w < 16; ++w) m = fmaxf(m, red[w * NH + tid]);
    mfin[tid] = m;
  }
  __syncthreads();

  // phase B: chunked exp + weighted accumulation; thread = (head, dim)
  int h = tid >> 7;        // 0..3
  int d = tid & 127;       // 0..127
  float acc = 0.0f;
  float denacc = 0.0f;     // meaningful for tid < 4 (deterministic serial sum)
  for (int p0 = 0; p0 < cnt; p0 += 32) {
    int pc = min(32, cnt - p0);
    if (tid < 128) {
      int pp = tid >> 2, hh = tid & 3;
      if (pp < pc) {
        int n = idx[off + p0 + pp];
        if (hh == 0) n_s[pp] = n;
        e_s[pp * NH + hh] = expf(scores[(size_t)n * NH + hh] - mfin[hh]);
      }
    }
    __syncthreads();
    if (tid < NH) for (int pp = 0; pp < pc; ++pp) denacc += e_s[pp * NH + tid];
    for (int pp = 0; pp < pc; ++pp)
      acc = fmaf(e_s[pp * NH + h], x[(size_t)n_s[pp] * D128 + d], acc);
    __syncthreads();
  }
  y[(size_t)r * (NH * D128) + tid] = acc;     // tid == h*128+d
  if (tid < NH) denom[r * NH + tid] = denacc;
}

// ---------------------------------------------------------------- pooled -> O = q + Wv_h@y/denom + bv
__global__ void k_pooled(const float* __restrict__ y, const float* __restrict__ denom,
                         const float* __restrict__ Wv, const float* __restrict__ bv,
                         const float* __restrict__ qv, float* __restrict__ O, int R) {
  int r = blockIdx.x;
  int j = threadIdx.x;                       // 128 threads
  if (r >= R) { O[(size_t)r * D128 + j] = 0.0f; return; }  // pad rows for WMMA tiles
  int h = j >> 5;
  const float* yr = y + (size_t)r * (NH * D128) + h * D128;
  const float* wr = Wv + (size_t)j * D128;
  float acc = 0.0f;
  for (int d = 0; d < D128; ++d) acc = fmaf(wr[d], yr[d], acc);
  float den = denom[r * NH + h];
  if (!(den > 0.0f)) den = 1.0f;
  O[(size_t)r * D128 + j] = qv[j] + acc / den + bv[j];
}

// ---------------------------------------------------------------- f32 WMMA 16x16x4 GEMM: out = [resid + relu(] A@W.T [+ bias )]
__global__ void k_gemm_wmma(const float* __restrict__ Amat, const float* __restrict__ Wmat,
                            const float* __restrict__ bias, const float* __restrict__ resid,
                            float* __restrict__ out, int relu_mode) {
  int bm = blockIdx.x;            // row tile
  int bn = blockIdx.y;            // col tile (0..7)
  int lane = threadIdx.x;         // 32 threads, EXEC all ones
  int half = lane >> 4;
  int l15  = lane & 15;
  const float* arow = Amat + (size_t)(bm * 16 + l15) * D128;  // A row (M = l15)
  const float* wrow = Wmat + (size_t)(bn * 16 + l15) * D128;  // W row n -> B col
  v8f c = {0.f, 0.f, 0.f, 0.f, 0.f, 0.f, 0.f, 0.f};
  for (int k0 = 0; k0 < D128; k0 += 4) {
    int kk = k0 + half * 2;
    v2f a; a[0] = arow[kk]; a[1] = arow[kk + 1];   // A 16x4: VGPR0={K0|K2}, VGPR1={K1|K3}
    v2f b; b[0] = wrow[kk]; b[1] = wrow[kk + 1];   // B 4x16: VGPR0={K0|K2}, VGPR1={K1|K3}
    c = __builtin_amdgcn_wmma_f32_16x16x4_f32(false, a, false, b, (short)0, c, false, false);
  }
  int col = bn * 16 + l15;
#pragma unroll
  for (int g = 0; g < 8; ++g) {                     // C/D: VGPR g -> rows g, g+8
    int row = bm * 16 + g + half * 8;
    float v = c[g];
    if (relu_mode) {
      v += bias[col];
      v = v > 0.0f ? v : 0.0f;
      v += resid[(size_t)row * D128 + col];
    }
    out[(size_t)row * D128 + col] = v;
  }
}

// ---------------------------------------------------------------- GCN degree / dis
__global__ void k_deg(const int* __restrict__ adj, int E, float* __restrict__ deg) {
  int e = blockIdx.x * blockDim.x + threadIdx.x;
  if (e >= E) return;
  atomicAdd(&deg[adj[E + e]], 1.0f);           // dst row
}
__global__ void k_dis(const float* __restrict__ deg, float* __restrict__ dis, int R) {
  int r = blockIdx.x * blockDim.x + threadIdx.x;
  if (r >= R) return;
  dis[r] = rsqrtf(fmaxf(deg[r], 1e-12f));
}

// ---------------------------------------------------------------- edge scatter: acc[dst] += norm*h[src]
__global__ void k_edges(const int* __restrict__ adj, int E, const float* __restrict__ dis,
                        const float* __restrict__ hbuf, float* __restrict__ acc) {
  int t = blockIdx.x * blockDim.x + threadIdx.x;
  if (t >= E * D128) return;
  int e = t >> 7;
  int d = t & 127;
  int s  = adj[e];
  int dt = adj[E + e];
  float nrm = dis[s] * dis[dt];
  atomicAdd(&acc[(size_t)dt * D128 + d], nrm * hbuf[(size_t)s * D128 + d]);
}

// ---------------------------------------------------------------- final: self loop + bg + PReLU
__global__ void k_final(const float* __restrict__ acc, const float* __restrict__ hbuf,
                        const float* __restrict__ dis, const float* __restrict__ bg,
                        const float* __restrict__ pw, float* __restrict__ out, int R) {
  int t = blockIdx.x * blockDim.x + threadIdx.x;
  if (t >= R * D128) return;
  int r = t >> 7;
  int d = t & 127;
  float ds = dis[r];
  float v = acc[t] + hbuf[t] * ds * ds + bg[d];
  out[t] = v > 0.0f ? v : pw[d] * v;
}

// ---------------------------------------------------------------- launch
extern "C" void kernel_launch(void* const* d_in, const int* in_sizes, int n_in,
                              void* d_out, int out_size, void* d_ws, size_t ws_size,
                              hipStream_t stream) {
  const float* x    = (const float*)d_in[0];
  const int*   zone = (const int*)d_in[1];
  const int*   adj  = (const int*)d_in[2];
  const float* S    = (const float*)d_in[3];
  const float* Wq   = (const float*)d_in[4];
  const float* bq   = (const float*)d_in[5];
  const float* Wk   = (const float*)d_in[6];
  const float* bk   = (const float*)d_in[7];
  const float* Wv   = (const float*)d_in[8];
  const float* bv   = (const float*)d_in[9];
  const float* Wo   = (const float*)d_in[10];
  const float* bo   = (const float*)d_in[11];
  const float* Wg   = (const float*)d_in[12];
  const float* bg   = (const float*)d_in[13];
  const float* pw   = (const float*)d_in[14];
  float* out = (float*)d_out;

  const int N = in_sizes[0] / D128;
  const int E = in_sizes[2] / 2;
  const int R = out_size / D128;
  const int M16 = (R + 15) & ~15;            // padded rows for 16x16 WMMA tiles

  // bump allocator over workspace
  char* wsb = (char*)d_ws;
  size_t pos = 0;
  auto alloc = [&](size_t bytes) -> void* {
    void* p = wsb + pos;
    pos = (pos + bytes + 255) & ~(size_t)255;
    return p;
  };
  float* qv      = (float*)alloc(D128 * 4);
  float* Av      = (float*)alloc(NH * D128 * 4);
  float* cv      = (float*)alloc(NH * 4);
  float* scores  = (float*)alloc((size_t)N * NH * 4);
  int*   counts  = (int*)alloc((size_t)R * 4);
  int*   offsets = (int*)alloc((size_t)(R + 1) * 4);
  int*   cursor  = (int*)alloc((size_t)R * 4);
  int*   idx     = (int*)alloc((size_t)N * 4);
  float* denom   = (float*)alloc((size_t)R * NH * 4);
  float* y       = (float*)alloc((size_t)R * NH * D128 * 4);
  float* Obuf    = (float*)alloc((size_t)M16 * D128 * 4);
  float* O2      = (float*)alloc((size_t)M16 * D128 * 4);
  float* Hbuf    = (float*)alloc((size_t)M16 * D128 * 4);
  float* deg     = (float*)alloc((size_t)R * 4);
  float* dis     = (float*)alloc((size_t)R * 4);
  float* acc     = (float*)alloc((size_t)R * D128 * 4);
  (void)ws_size; (void)n_in;

  k_init<<<(R * D128 + 255) / 256, 256, 0, stream>>>(counts, cursor, deg, acc, R);
  k_prep<<<1, 128, 0, stream>>>(S, Wq, bq, Wk, bk, qv, Av, cv);
  k_scores<<<(N + 255) / 256, 256, 0, stream>>>(x, zone, Av, cv, scores, counts, N);
  k_scan<<<1, 1024, 0, stream>>>(counts, offsets, R);
  k_scatter<<<(N + 255) / 256, 256, 0, stream>>>(zone, offsets, cursor, idx, N);
  k_region<<<R, K5T, 0, stream>>>(x, scores, idx, offsets, y, denom);
  k_pooled<<<M16, 128, 0, stream>>>(y, denom, Wv, bv, qv, Obuf, R);
  k_gemm_wmma<<<dim3(M16 / 16, D128 / 16), 32, 0, stream>>>(Obuf, Wo, bo, Obuf, O2, 1);
  k_deg<<<(E + 255) / 256, 256, 0, stream>>>(adj, E, deg);
  k_dis<<<(R + 255) / 256, 256, 0, stream>>>(deg, dis, R);
  k_gemm_wmma<<<dim3(M16 / 16, D128 / 16), 32, 0, stream>>>(O2, Wg, bo, O2, Hbuf, 0);
  k_edges<<<(E * D128 + 255) / 256, 256, 0, stream>>>(adj, E, dis, Hbuf, acc);
  k_final<<<(R * D128 + 255) / 256, 256, 0, stream>>>(acc, Hbuf, dis, bg, pw, out, R);
}